// MultiScaleAttention_18227841204880
// MI455X (gfx1250) — compile-verified
//
#include <hip/hip_runtime.h>
#include <math.h>

typedef __attribute__((ext_vector_type(2))) float v2f;
typedef __attribute__((ext_vector_type(4))) float v4f;
typedef __attribute__((ext_vector_type(8))) float v8f;

#define NB   8
#define NT   1569      // q tokens (8*14*14+1)
#define NC   768
#define NHD  8
#define HD   96
#define NKV  393       // pooled k/v tokens (8*7*7+1)
#define T0_  8
#define H0_  14
#define W0_  14
#define SCALE_F 0.10206207261596577f   // 96^-0.5

// ---- CDNA5 async global->LDS path (guarded; falls back to plain copy) ----
#ifdef __has_builtin
#if __has_builtin(__builtin_amdgcn_global_load_async_to_lds_b32)
#define HAVE_ASYNC_LDS 1
#endif
#endif
#ifndef HAVE_ASYNC_LDS
#define HAVE_ASYNC_LDS 0
#endif

// Builtin signature (from probe diagnostics): (__device__ int*, __shared__ int*, imm, imm).
// Numbered AS(1)/AS(3) share the target address spaces of __device__/__shared__.
#define GPTR(p) ((__attribute__((address_space(1))) int*)(float*)(p))
#define LPTR(p) ((__attribute__((address_space(3))) int*)(p))

static __device__ __forceinline__ void wait_async0() {
#if HAVE_ASYNC_LDS
#if __has_builtin(__builtin_amdgcn_s_wait_asynccnt)
    __builtin_amdgcn_s_wait_asynccnt(0);
#else
    asm volatile("s_wait_asynccnt 0x0" ::: "memory");
#endif
#endif
}

static __device__ __forceinline__ v8f wmma_f32(v2f a, v2f b, v8f c) {
    // V_WMMA_F32_16X16X4_F32: D = A(16x4) * B(4x16) + C, fp32 matrix pipe
    return __builtin_amdgcn_wmma_f32_16x16x4_f32(
        false, a, false, b, (short)0, c, false, false);
}

// ---------------------------------------------------------------------------
// LDS double-buffered GEMM with bias: C[M,N] = A[M,K] @ B[K,N] + bias[N]
// Block tile 128x128x16, 256 threads = 8 waves (2Mx4N), wave tile 64x32.
// A staged [128][20] row-major; B staged transposed [128][20] so both WMMA
// fragment reads are contiguous ds_load_b64.  Requires N % 128 == 0.
// ---------------------------------------------------------------------------
#define BM 128
#define BN 128
#define BK 16
#define LDA 20
#define LDB 20

__global__ __launch_bounds__(256) void gemm_bias_kernel(
    const float* __restrict__ A, const float* __restrict__ Bm,
    const float* __restrict__ bias, float* __restrict__ Cm,
    int M, int K, int Nn)
{
    __shared__ float As[2][BM * LDA];
    __shared__ float Bs[2][BN * LDB];

    const int tid  = threadIdx.x;
    const int lane = tid & 31, wave = tid >> 5;
    const int lo   = lane & 15, hi = lane >> 4;
    const int waveM = wave >> 2, waveN = wave & 3;      // 2 x 4
    const int rowBase = blockIdx.y * BM;
    const int colBase = blockIdx.x * BN;

    // accumulators: 4 M-subtiles x 2 N-subtiles of 16x16
    v8f acc[4][2];
#pragma unroll
    for (int n = 0; n < 2; ++n) {
        float bz = bias[colBase + waveN * 32 + n * 16 + lo];
#pragma unroll
        for (int m = 0; m < 4; ++m)
#pragma unroll
            for (int r = 0; r < 8; ++r) acc[m][n][r] = bz;
    }

    // global-load index mapping (two b128 loads each for A and B per tile)
    const int afRow = tid >> 2;              // 0..63   (second load: +64)
    const int afQ   = (tid & 3) * 4;         // k sub-offset 0/4/8/12
    const int bfK   = tid >> 5;              // 0..7    (second load: +8)
    const int bfC   = (tid & 31) * 4;        // col sub-offset

    int ar0 = rowBase + afRow;       if (ar0 > M - 1) ar0 = M - 1;
    int ar1 = rowBase + afRow + 64;  if (ar1 > M - 1) ar1 = M - 1;
    const float* Ap0 = A + (size_t)ar0 * K + afQ;
    const float* Ap1 = A + (size_t)ar1 * K + afQ;
    const float* Bp0 = Bm + (size_t)bfK * Nn + colBase + bfC;
    const float* Bp1 = Bm + (size_t)(bfK + 8) * Nn + colBase + bfC;

    v4f aReg0, aReg1, bReg0, bReg1;

    // prologue: tile 0
    aReg0 = *(const v4f*)(Ap0);
    aReg1 = *(const v4f*)(Ap1);
    bReg0 = *(const v4f*)(Bp0);
    bReg1 = *(const v4f*)(Bp1);
    *(v4f*)(&As[0][afRow * LDA + afQ])        = aReg0;
    *(v4f*)(&As[0][(afRow + 64) * LDA + afQ]) = aReg1;
#pragma unroll
    for (int i = 0; i < 4; ++i) {
        Bs[0][(bfC + i) * LDB + bfK]     = bReg0[i];
        Bs[0][(bfC + i) * LDB + bfK + 8] = bReg1[i];
    }
    __syncthreads();

    const int nTiles = K / BK;                           // 48
    for (int t = 0; t < nTiles; ++t) {
        const int buf = t & 1;
        if (t + 1 < nTiles) {                            // prefetch tile t+1
            size_t ko = (size_t)(t + 1) * BK;
            aReg0 = *(const v4f*)(Ap0 + ko);
            aReg1 = *(const v4f*)(Ap1 + ko);
            bReg0 = *(const v4f*)(Bp0 + ko * Nn);
            bReg1 = *(const v4f*)(Bp1 + ko * Nn);
        }

        // compute from LDS buffer `buf`
        const float* as = &As[buf][(waveM * 64 + lo) * LDA + 2 * hi];
        const float* bs = &Bs[buf][(waveN * 32 + lo) * LDB + 2 * hi];
#pragma unroll
        for (int kk = 0; kk < BK; kk += 4) {
            v2f af[4], bf[2];
#pragma unroll
            for (int m = 0; m < 4; ++m)
                af[m] = *(const v2f*)(as + m * 16 * LDA + kk);
#pragma unroll
            for (int n = 0; n < 2; ++n)
                bf[n] = *(const v2f*)(bs + n * 16 * LDB + kk);
#pragma unroll
            for (int m = 0; m < 4; ++m)
#pragma unroll
                for (int n = 0; n < 2; ++n)
                    acc[m][n] = wmma_f32(af[m], bf[n], acc[m][n]);
        }

        if (t + 1 < nTiles) {
            __syncthreads();                             // all reads of nxt done
            const int nxt = buf ^ 1;
            *(v4f*)(&As[nxt][afRow * LDA + afQ])        = aReg0;
            *(v4f*)(&As[nxt][(afRow + 64) * LDA + afQ]) = aReg1;
#pragma unroll
            for (int i = 0; i < 4; ++i) {
                Bs[nxt][(bfC + i) * LDB + bfK]     = bReg0[i];
                Bs[nxt][(bfC + i) * LDB + bfK + 8] = bReg1[i];
            }
            __syncthreads();
        }
    }

    // store
#pragma unroll
    for (int m = 0; m < 4; ++m) {
#pragma unroll
        for (int r = 0; r < 8; ++r) {
            int row = rowBase + waveM * 64 + m * 16 + r + 8 * hi;
            if (row < M) {
#pragma unroll
                for (int n = 0; n < 2; ++n)
                    Cm[(size_t)row * Nn + colBase + waveN * 32 + n * 16 + lo] =
                        acc[m][n][r];
            }
        }
    }
}

// ---------------------------------------------------------------------------
// Depthwise conv3d (3x3x3, pad 1) + LayerNorm over D=96.
// One wave per (b, head, out_token); 3 channels per lane.
// qkv layout: [B, N, 3, NH, D]
// ---------------------------------------------------------------------------
__global__ __launch_bounds__(256) void pool_norm_kernel(
    const float* __restrict__ qkv, int j, const float* __restrict__ cw,
    const float* __restrict__ g, const float* __restrict__ beta,
    float* __restrict__ outp, int st, int oh, int ow)
{
    const int lane = threadIdx.x & 31;
    int wv = (int)((blockIdx.x * blockDim.x + threadIdx.x) >> 5);
    const int Nout = T0_ * oh * ow + 1;
    if (wv >= NB * NHD * Nout) return;

    int tok  = wv % Nout;
    int tmp  = wv / Nout;
    int head = tmp % NHD;
    int b    = tmp / NHD;

    const size_t srcBase =
        (size_t)b * NT * (3 * NC) + (size_t)j * NC + (size_t)head * HD;

    float val[3];
    if (tok == 0) {
#pragma unroll
        for (int s = 0; s < 3; ++s) val[s] = qkv[srcBase + lane + 32 * s];
    } else {
        int sp = tok - 1;
        int to = sp / (oh * ow);
        int rem = sp % (oh * ow);
        int ho = rem / ow, wo = rem % ow;
#pragma unroll
        for (int s = 0; s < 3; ++s) {
            int c = lane + 32 * s;
            float acc = 0.f;
            for (int kt = 0; kt < 3; ++kt) {
                int it = to - 1 + kt;                 // t-stride is always 1
                if (it < 0 || it >= T0_) continue;
                for (int kh = 0; kh < 3; ++kh) {
                    int ih = ho * st - 1 + kh;
                    if (ih < 0 || ih >= H0_) continue;
                    for (int kw = 0; kw < 3; ++kw) {
                        int iw = wo * st - 1 + kw;
                        if (iw < 0 || iw >= W0_) continue;
                        int nin = 1 + (it * H0_ + ih) * W0_ + iw;
                        acc += qkv[srcBase + (size_t)nin * (3 * NC) + c] *
                               cw[c * 27 + kt * 9 + kh * 3 + kw];
                    }
                }
            }
            val[s] = acc;
        }
    }

    float s1 = val[0] + val[1] + val[2];
    float s2 = val[0] * val[0] + val[1] * val[1] + val[2] * val[2];
#pragma unroll
    for (int m = 1; m < 32; m <<= 1) {
        s1 += __shfl_xor(s1, m, 32);
        s2 += __shfl_xor(s2, m, 32);
    }
    float mu  = s1 * (1.f / 96.f);
    float var = s2 * (1.f / 96.f) - mu * mu;
    float rs  = rsqrtf(var + 1e-5f);

    size_t ob = (((size_t)b * NHD + head) * Nout + tok) * HD;
#pragma unroll
    for (int s = 0; s < 3; ++s) {
        int c = lane + 32 * s;
        outp[ob + c] = (val[s] - mu) * rs * g[c] + beta[c];
    }
}

// ---------------------------------------------------------------------------
// Flash attention per (b, head). Block = 8 waves; each wave owns 16 q-rows
// (128 q rows/block).  k-tile and a TRANSPOSED v-tile are staged in LDS with
// CDNA5 async global->LDS copies (ASYNCcnt) when available; QK^T and P@V run
// on fp32 WMMA with contiguous ds_load_b64 fragment reads.  Online softmax.
// ---------------------------------------------------------------------------
#define LDK 98     // k tile row pitch  [16][98]
#define LDV 18     // vT tile row pitch [96][18]
#define LDP 18     // P  tile row pitch [16][18] per wave

__global__ __launch_bounds__(256) void attn_kernel(
    const float* __restrict__ qp, const float* __restrict__ kp,
    const float* __restrict__ vp, float* __restrict__ outh)
{
    __shared__ float ktile[16 * LDK];
    __shared__ float vtile[HD * LDV];
    __shared__ float ptile[8 * 16 * LDP];

    const int tid  = threadIdx.x;
    const int lane = tid & 31;
    const int wave = tid >> 5;
    const int lo = lane & 15, hi = lane >> 4;
    const int pair = blockIdx.y;
    const int bb = pair >> 3, hh = pair & 7;
    const int qbase = (blockIdx.x * 8 + wave) * 16;   // may exceed NT: redundant work

    const float* qb = qp + ((size_t)bb * NHD + hh) * NT  * HD;
    const float* kb = kp + ((size_t)bb * NHD + hh) * NKV * HD;
    const float* vb = vp + ((size_t)bb * NHD + hh) * NKV * HD;
    float* myp = ptile + wave * (16 * LDP);

    int qrow = qbase + lo;
    if (qrow > NT - 1) qrow = NT - 1;
    v2f qa[24];                                        // whole 16x96 q tile
#pragma unroll
    for (int s = 0; s < 24; ++s)
        qa[s] = *(const v2f*)(qb + (size_t)qrow * HD + 4 * s + 2 * hi);

    v8f o[6] = {};
    float mrow[8], lrow[8];
#pragma unroll
    for (int r = 0; r < 8; ++r) { mrow[r] = -1e30f; lrow[r] = 0.f; }

    for (int jt = 0; jt < 25; ++jt) {                  // ceil(393/16)
        const int jbase = jt * 16;

        __syncthreads();                               // previous tile fully read
#pragma unroll
        for (int i = 0; i < 6; ++i) {                  // 1536 elems / 256 thr
            int e = tid + i * 256;
            int j = e / HD, d = e % HD;
            int row = jbase + j;
            if (row > NKV - 1) row = NKV - 1;          // masked later
#if HAVE_ASYNC_LDS
            __builtin_amdgcn_global_load_async_to_lds_b32(
                GPTR(kb + (size_t)row * HD + d), LPTR(&ktile[j * LDK + d]), 0, 0);
            __builtin_amdgcn_global_load_async_to_lds_b32(
                GPTR(vb + (size_t)row * HD + d), LPTR(&vtile[d * LDV + j]), 0, 0);
#else
            ktile[j * LDK + d] = kb[(size_t)row * HD + d];
            vtile[d * LDV + j] = vb[(size_t)row * HD + d];
#endif
        }
        wait_async0();                                 // this wave's copies done
        __syncthreads();                               // all waves' copies done

        // ---- QK^T (16 x 16 scores) ----
        v8f sc = {};
        const float* kr = ktile + lo * LDK + 2 * hi;
#pragma unroll
        for (int s = 0; s < 24; ++s) {
            v2f kf = *(const v2f*)(kr + 4 * s);        // B[k][j] = k[j][k]
            sc = wmma_f32(qa[s], kf, sc);
        }

        const bool valid = (jbase + lo) < NKV;
#pragma unroll
        for (int r = 0; r < 8; ++r)
            sc[r] = valid ? sc[r] * SCALE_F : -1e30f;

        // ---- online softmax ----
#pragma unroll
        for (int r = 0; r < 8; ++r) {
            float rm = sc[r];
#pragma unroll
            for (int m = 1; m < 16; m <<= 1) rm = fmaxf(rm, __shfl_xor(rm, m, 32));
            float mnew = fmaxf(mrow[r], rm);
            float corr = __expf(mrow[r] - mnew);
            float p    = __expf(sc[r] - mnew);
            float rs   = p;
#pragma unroll
            for (int m = 1; m < 16; m <<= 1) rs += __shfl_xor(rs, m, 32);
            lrow[r] = lrow[r] * corr + rs;
            mrow[r] = mnew;
#pragma unroll
            for (int d = 0; d < 6; ++d) o[d][r] *= corr;
            myp[(r + 8 * hi) * LDP + lo] = p;          // P[m][j] (private region)
        }
        __builtin_amdgcn_wave_barrier();               // DS in-order per wave

        // ---- P @ V ----
#pragma unroll
        for (int dc = 0; dc < 6; ++dc) {
            const float* vr = vtile + (dc * 16 + lo) * LDV + 2 * hi;
#pragma unroll
            for (int kk = 0; kk < 16; kk += 4) {
                v2f a = *(const v2f*)(myp + lo * LDP + kk + 2 * hi);
                v2f b = *(const v2f*)(vr + kk);        // vT: contiguous over j
                o[dc] = wmma_f32(a, b, o[dc]);
            }
        }
        __builtin_amdgcn_wave_barrier();
    }

#pragma unroll
    for (int r = 0; r < 8; ++r) {
        int row = qbase + r + 8 * hi;
        if (row < NT) {
            float inv = 1.f / lrow[r];
            size_t ob = ((size_t)bb * NT + row) * NC + (size_t)hh * HD;
#pragma unroll
            for (int dc = 0; dc < 6; ++dc)
                outh[ob + dc * 16 + lo] = o[dc][r] * inv;
        }
    }
}

// ---------------------------------------------------------------------------
extern "C" void kernel_launch(void* const* d_in, const int* in_sizes, int n_in,
                              void* d_out, int out_size, void* d_ws, size_t ws_size,
                              hipStream_t stream) {
    (void)in_sizes; (void)n_in; (void)out_size; (void)ws_size;

    const float* x     = (const float*)d_in[0];
    const float* Wqkv  = (const float*)d_in[1];
    const float* bqkv  = (const float*)d_in[2];
    const float* Wproj = (const float*)d_in[3];
    const float* bproj = (const float*)d_in[4];
    const float* cwq   = (const float*)d_in[5];
    const float* cwk   = (const float*)d_in[6];
    const float* cwv   = (const float*)d_in[7];
    const float* lnqg  = (const float*)d_in[8];
    const float* lnqb  = (const float*)d_in[9];
    const float* lnkg  = (const float*)d_in[10];
    const float* lnkb  = (const float*)d_in[11];
    const float* lnvg  = (const float*)d_in[12];
    const float* lnvb  = (const float*)d_in[13];

    const int M = NB * NT;                       // 12552
    float* ws   = (float*)d_ws;
    float* qkv  = ws;                                        // M * 2304
    float* qp   = qkv + (size_t)M * 3 * NC;                  // 64*1569*96
    float* kpp  = qp  + (size_t)NB * NHD * NT  * HD;         // 64*393*96
    float* vpp  = kpp + (size_t)NB * NHD * NKV * HD;         // 64*393*96
    float* outh = vpp + (size_t)NB * NHD * NKV * HD;         // M * 768

    dim3 blk(256);

    // 1) qkv = x @ W_qkv + b_qkv   (grid: 2304/128 x ceil(12552/128))
    gemm_bias_kernel<<<dim3((3 * NC) / BN, (M + BM - 1) / BM), blk, 0, stream>>>(
        x, Wqkv, bqkv, qkv, M, NC, 3 * NC);

    // 2) pool + LayerNorm for q (stride 1) and k/v (stride 2)
    int wavesQ = NB * NHD * NT;
    pool_norm_kernel<<<dim3((wavesQ + 7) / 8), blk, 0, stream>>>(
        qkv, 0, cwq, lnqg, lnqb, qp, 1, 14, 14);
    int wavesK = NB * NHD * NKV;
    pool_norm_kernel<<<dim3((wavesK + 7) / 8), blk, 0, stream>>>(
        qkv, 1, cwk, lnkg, lnkb, kpp, 2, 7, 7);
    pool_norm_kernel<<<dim3((wavesK + 7) / 8), blk, 0, stream>>>(
        qkv, 2, cwv, lnvg, lnvb, vpp, 2, 7, 7);

    // 3) attention (flash-style, WMMA fp32, async k/vT staging)
    attn_kernel<<<dim3((NT + 127) / 128, NB * NHD), blk, 0, stream>>>(
        qp, kpp, vpp, outh);

    // 4) out = outh @ W_proj + b_proj
    gemm_bias_kernel<<<dim3(NC / BN, (M + BM - 1) / BM), blk, 0, stream>>>(
        outh, Wproj, bproj, (float*)d_out, M, NC, NC);
}